// Vibes_82892868812891
// MI455X (gfx1250) — compile-verified
//
#include <hip/hip_runtime.h>
#include <math.h>

#define L 128
#define B 32
#define D 64
#define HID 32
#define NROWS 4096   /* L*B */
#define NSTEPS 10
#define DTc 0.1f

typedef __attribute__((ext_vector_type(2))) float v2f;
typedef __attribute__((ext_vector_type(8))) float v8f;

// ---- workspace layout (in floats) ----
#define OFF_S      0            // suffix sums: 128*64*32 = 262144
#define OFF_G      262144       // g: 4096*32   = 131072
#define OFF_DIAG   393216       // diag: 4096*64 = 262144
#define OFF_ERR    655360       // err: 4096
#define OFF_SCORE  659456       // score: 4096*64 = 262144
#define OFF_PART   921600       // wave partials: 1024
#define OFF_SNORM  922624       // 1
#define OFF_ABSP   922640       // block partials: 1024

// ---------------- RNG: deterministic counter hash -> Box-Muller ------------
__device__ inline unsigned hashu(unsigned x) {
    x ^= x >> 16; x *= 0x7feb352dU;
    x ^= x >> 15; x *= 0x846ca68bU;
    x ^= x >> 16; return x;
}
__device__ inline float rngnorm(int s, int idx) {
    unsigned h1 = hashu((unsigned)idx * 0x9E3779B9U + (unsigned)s * 0x85EBCA77U + 0x12345687U);
    unsigned h2 = hashu(h1 ^ 0x68bc21ebU);
    float u1 = ((float)(h1 >> 8) + 0.5f) * (1.0f / 16777216.0f);
    float u2 = (float)(h2 >> 8) * (1.0f / 16777216.0f);
    return sqrtf(-2.0f * __logf(u1)) * __cosf(6.2831853071795864f * u2);
}

// ---------------- zero diag + err ------------------------------------------
__global__ __launch_bounds__(256) void k_zero(float* ws) {
    int tid = blockIdx.x * 256 + threadIdx.x;       // 266240 threads exactly
    ws[OFF_DIAG + tid] = 0.0f;                      // covers diag (262144) then err (4096)
}

// ---------------- suffix sums of W1 over t ---------------------------------
// S[j,d,k] = sum_{t>j} W1[t*64+d, k]
__global__ __launch_bounds__(256) void k_suffix(const float* __restrict__ W1, float* ws) {
    int tid = blockIdx.x * 256 + threadIdx.x;       // 2048 threads: (d,k)
    int k = tid & 31, d = tid >> 5;
    float* S = ws + OFF_S;
    float acc = 0.0f;
    S[((127 * 64 + d) << 5) + k] = 0.0f;
    for (int j = 126; j >= 0; --j) {
        acc += W1[(((j + 1) * 64 + d) << 5) + k];
        S[((j * 64 + d) << 5) + k] = acc;
    }
}

// ---------------- g0[jb,k] = sum_d emb[b,j,d] * S[j,d,k] -------------------
__global__ __launch_bounds__(256) void k_g0(const float* __restrict__ emb, float* ws) {
    int tid = blockIdx.x * 256 + threadIdx.x;       // 131072 threads: (r,k)
    int r = tid >> 5, k = tid & 31;
    int j = r >> 5, b = r & 31;                     // row index jb = j*B + b
    const float* S = ws + OFF_S;
    const float* e = emb + ((b * 128 + j) << 6);
    float acc = 0.0f;
    for (int d = 0; d < 64; ++d)
        acc = fmaf(e[d], S[((j * 64 + d) << 5) + k], acc);
    ws[OFF_G + (r << 5) + k] = acc;
}

// ---------------- per-step: score = relu(g + t*w1last + b1) @ W2 + b2 ------
// One wave per 16x16 output tile; K=32 split into 8 chunks of 4 (f32 WMMA).
__global__ __launch_bounds__(256) void k_score(const float* __restrict__ W1,
                                               const float* __restrict__ b1,
                                               const float* __restrict__ W2,
                                               const float* __restrict__ b2,
                                               float* ws, float tval) {
    const float* g      = ws + OFF_G;
    const float* w1last = W1 + 8192 * 32;
    float*       score  = ws + OFF_SCORE;
    float*       part   = ws + OFF_PART;

    int wid  = (blockIdx.x * 256 + threadIdx.x) >> 5;   // 0..1023
    int lane = threadIdx.x & 31;
    int nt = wid & 3, rt = wid >> 2;                    // N tile, M tile
    int mrow  = rt * 16 + (lane & 15);                  // A row for this lane
    int col   = nt * 16 + (lane & 15);                  // B/C column for this lane
    int khalf = (lane & 16) ? 2 : 0;

    v8f c = {};
#pragma unroll
    for (int kc = 0; kc < 8; ++kc) {
        int k0 = kc * 4 + khalf;
        float h0 = fmaxf(g[(mrow << 5) + k0]     + tval * w1last[k0]     + b1[k0],     0.0f);
        float h1 = fmaxf(g[(mrow << 5) + k0 + 1] + tval * w1last[k0 + 1] + b1[k0 + 1], 0.0f);
        v2f a; a.x = h0; a.y = h1;
        v2f bb; bb.x = W2[(k0 << 6) + col]; bb.y = W2[((k0 + 1) << 6) + col];
        c = __builtin_amdgcn_wmma_f32_16x16x4_f32(false, a, false, bb, (short)0, c, false, false);
    }
    float bias  = b2[col];
    int   rbase = rt * 16 + ((lane & 16) ? 8 : 0);
    float ss = 0.0f;
#pragma unroll
    for (int v = 0; v < 8; ++v) {
        float val = c[v] + bias;
        score[((rbase + v) << 6) + col] = val;
        ss += val * val;
    }
#pragma unroll
    for (int off = 16; off; off >>= 1) ss += __shfl_xor(ss, off);
    if (lane == 0) part[wid] = ss;
}

// ---------------- deterministic snorm reduction ----------------------------
__global__ __launch_bounds__(256) void k_snorm(float* ws) {
    __shared__ float sm[256];
    const float* part = ws + OFF_PART;
    int t = threadIdx.x;
    float s = part[t] + part[t + 256] + part[t + 512] + part[t + 768];
    sm[t] = s; __syncthreads();
    for (int off = 128; off; off >>= 1) {
        if (t < off) sm[t] += sm[t + off];
        __syncthreads();
    }
    if (t == 0) ws[OFF_SNORM] = sqrtf(sm[0]);
}

// ---------------- per-step update: dx, diag, err, incremental g ------------
__global__ __launch_bounds__(256) void k_update(const float* __restrict__ W1,
                                                float* ws, int s) {
    float* g     = ws + OFF_G;
    float* diag  = ws + OFF_DIAG;
    float* err   = ws + OFF_ERR;
    const float* score = ws + OFF_SCORE;

    int lane = threadIdx.x & 31;
    int r    = blockIdx.x * 8 + (threadIdx.x >> 5);     // one wave per row
    float snorm = ws[OFF_SNORM];
    float sd = snorm * 0.31622776601683794f;            // sqrt(DT)
    int d0 = lane, d1 = lane + 32;

    float dx0 = 0.05f * score[(r << 6) + d0] + sd * rngnorm(s, (r << 6) + d0);
    float dx1 = 0.05f * score[(r << 6) + d1] + sd * rngnorm(s, (r << 6) + d1);
    float dg0 = diag[(r << 6) + d0] + dx0; diag[(r << 6) + d0] = dg0;
    float dg1 = diag[(r << 6) + d1] + dx1; diag[(r << 6) + d1] = dg1;

    if ((r & 127) == (r >> 5)) {                        // rows feeding preds/error
        float e = dg0 * dg0 + dg1 * dg1;
#pragma unroll
        for (int off = 16; off; off >>= 1) e += __shfl_xor(e, off);
        if (lane == 0) err[r] += e * (1.0f / 64.0f);
    }

    const float* w1b = W1 + (((r & 127) << 6) << 5);    // W1[(r%128)*64 + d, k]
    float acc = g[(r << 5) + lane];
#pragma unroll
    for (int d = 0; d < 32; ++d)
        acc = fmaf(__shfl(dx0, d), w1b[(d << 5) + lane], acc);
#pragma unroll
    for (int d = 0; d < 32; ++d)
        acc = fmaf(__shfl(dx1, d), w1b[((d + 32) << 5) + lane], acc);
    g[(r << 5) + lane] = acc;
}

// ---------------- final gather + |diff| block partials ---------------------
__global__ __launch_bounds__(256) void k_final(const float* __restrict__ emb,
                                               float* ws, float* out) {
    __shared__ float sm[256];
    const float* diag = ws + OFF_DIAG;
    int tid = blockIdx.x * 256 + threadIdx.x;           // 262144 threads
    int b = tid >> 13, i = (tid >> 6) & 127, d = tid & 63;
    int m = 32 * i + b;
    int t = m & 127, q = m >> 7;
    int r = 32 * i + q;
    float val = (i < t) ? emb[((q * 128 + i) << 6) + d] : 0.0f;
    if (t == (r & 127)) val += diag[(r << 6) + d];
    out[tid] = val;
    float diff = fabsf(emb[((b * 128 + i) << 6) + d] - val);
    sm[threadIdx.x] = diff; __syncthreads();
    for (int off = 128; off; off >>= 1) {
        if (threadIdx.x < off) sm[threadIdx.x] += sm[threadIdx.x + off];
        __syncthreads();
    }
    if (threadIdx.x == 0) ws[OFF_ABSP + blockIdx.x] = sm[0];
}

// ---------------- scalars: step_loss, sequence_loss ------------------------
__global__ __launch_bounds__(256) void k_scalars(float* ws, float* out) {
    __shared__ float sm[256];
    int t = threadIdx.x;
    const float* err  = ws + OFF_ERR;
    const float* absp = ws + OFF_ABSP;
    float se = 0.0f;
    for (int k = 0; k < 16; ++k) se += err[t + (k << 8)];
    sm[t] = se; __syncthreads();
    for (int off = 128; off; off >>= 1) { if (t < off) sm[t] += sm[t + off]; __syncthreads(); }
    float step_loss = sm[0]; __syncthreads();
    float sa = absp[t] + absp[t + 256] + absp[t + 512] + absp[t + 768];
    sm[t] = sa; __syncthreads();
    for (int off = 128; off; off >>= 1) { if (t < off) sm[t] += sm[t + off]; __syncthreads(); }
    if (t == 0) {
        out[262144] = step_loss;
        out[262145] = sm[0] * (1.0f / 262144.0f);
    }
}

extern "C" void kernel_launch(void* const* d_in, const int* in_sizes, int n_in,
                              void* d_out, int out_size, void* d_ws, size_t ws_size,
                              hipStream_t stream) {
    const float* emb = (const float*)d_in[0];
    const float* W1  = (const float*)d_in[1];
    const float* b1  = (const float*)d_in[2];
    const float* W2  = (const float*)d_in[3];
    const float* b2  = (const float*)d_in[4];
    float* out = (float*)d_out;
    float* ws  = (float*)d_ws;
    (void)in_sizes; (void)n_in; (void)out_size; (void)ws_size;

    k_zero  <<<1040, 256, 0, stream>>>(ws);
    k_suffix<<<   8, 256, 0, stream>>>(W1, ws);
    k_g0    <<< 512, 256, 0, stream>>>(emb, ws);
    for (int s = 0; s < NSTEPS; ++s) {
        float tval = (float)s * DTc;
        k_score <<<128, 256, 0, stream>>>(W1, b1, W2, b2, ws, tval);
        k_snorm <<<  1, 256, 0, stream>>>(ws);
        k_update<<<512, 256, 0, stream>>>(W1, ws, s);
    }
    k_final  <<<1024, 256, 0, stream>>>(emb, ws, out);
    k_scalars<<<   1, 256, 0, stream>>>(ws, out);
}